// DIoULoss_16406775071070
// MI455X (gfx1250) — compile-verified
//
#include <hip/hip_runtime.h>

// DIoU loss, mean reduction. Memory-bound: 64MB in, 4B out (~2.7us @ 23.3TB/s).
// Main loop: b128 coalesced loads + scalar f32 math, single fused IEEE divide:
//   diou = inter/uni - cdist/diag = (inter*diag - cdist*uni) / (uni*diag)
// Wave reduction: exact f32 via V_WMMA_F32_16X16X4_F32 ones-trick (CDNA5 path).

typedef __attribute__((ext_vector_type(2))) float v2f;
typedef __attribute__((ext_vector_type(8))) float v8f;

#define NBLOCKS  1024
#define NTHREADS 256
#define NWAVES   (NTHREADS / 32)

// Exact 32-lane f32 sum using one WMMA.
// A-matrix 16x4 f32 layout (wave32): lanes 0-15 hold K=0 (VGPR0), K=1 (VGPR1);
// lanes 16-31 hold K=2,3. With a = (v, 0) per lane and B = all-ones (4x16),
// D[m][n] = v(m) + v(m+16) for every column n.
// C/D layout: lane L, VGPR r holds M = r + 8*(L>=16), N = L%16. Summing the
// 8 D registers gives lanes 0-15: sum_{m=0..7} rowsum(m), lanes 16-31:
// sum_{m=8..15} rowsum(m). One xor-16 shuffle completes the wave total.
__device__ __forceinline__ float wave_reduce_wmma(float v) {
    v2f a; a.x = v;    a.y = 0.0f;
    v2f b; b.x = 1.0f; b.y = 1.0f;
    v8f c = {};
    c = __builtin_amdgcn_wmma_f32_16x16x4_f32(
            /*neg_a=*/false, a, /*neg_b=*/false, b,
            /*c_mod=*/(short)0, c, /*reuse_a=*/false, /*reuse_b=*/false);
    float s = c[0] + c[1] + c[2] + c[3] + c[4] + c[5] + c[6] + c[7];
    s += __shfl_xor(s, 16, 32);
    return s;  // all lanes hold the 32-lane sum
}

// Returns block total on thread 0 (undefined elsewhere). All threads must call.
__device__ __forceinline__ float block_reduce(float v, float* sdata) {
    float wsum = wave_reduce_wmma(v);
    const int wave = threadIdx.x >> 5;
    const int lane = threadIdx.x & 31;
    if (lane == 0) sdata[wave] = wsum;
    __syncthreads();
    float total = 0.0f;
    if (threadIdx.x == 0) {
#pragma unroll
        for (int w = 0; w < NWAVES; ++w) total += sdata[w];
    }
    return total;
}

__device__ __forceinline__ float diou_loss(float4 p, float4 t) {
    // intersection
    const float ltx = fmaxf(p.x, t.x), lty = fmaxf(p.y, t.y);
    const float rbx = fminf(p.z, t.z), rby = fminf(p.w, t.w);
    const float w = fmaxf(rbx - ltx, 0.0f);
    const float h = fmaxf(rby - lty, 0.0f);
    const float inter = w * h;
    // union
    const float area_p = (p.z - p.x) * (p.w - p.y);
    const float area_t = (t.z - t.x) * (t.w - t.y);
    const float uni = area_p + area_t - inter;
    // squared center distance
    const float dx = (p.x + p.z) * 0.5f - (t.x + t.z) * 0.5f;
    const float dy = (p.y + p.w) * 0.5f - (t.y + t.w) * 0.5f;
    const float cdist = dx * dx + dy * dy;
    // squared enclosing-box diagonal
    const float ex = fmaxf(p.z, t.z) - fminf(p.x, t.x);
    const float ey = fmaxf(p.w, t.w) - fminf(p.y, t.y);
    const float diag = ex * ex + ey * ey;
    // diou = inter/uni - cdist/diag, fused into one divide
    const float num = inter * diag - cdist * uni;
    const float den = uni * diag;
    return 1.0f - num / den;
}

__global__ void __launch_bounds__(NTHREADS)
diou_partial_kernel(const float4* __restrict__ pred,
                    const float4* __restrict__ tgt,
                    float* __restrict__ partials, int n) {
    __shared__ float sdata[NWAVES];
    float acc = 0.0f;
    const int stride = gridDim.x * blockDim.x;
    for (int i = blockIdx.x * blockDim.x + threadIdx.x; i < n; i += stride) {
        const float4 p = pred[i];   // global_load_b128, coalesced
        const float4 t = tgt[i];    // global_load_b128, coalesced
        acc += diou_loss(p, t);
    }
    const float total = block_reduce(acc, sdata);
    if (threadIdx.x == 0) partials[blockIdx.x] = total;
}

__global__ void __launch_bounds__(NTHREADS)
diou_final_kernel(const float* __restrict__ partials, int np,
                  float* __restrict__ out, float inv_n) {
    __shared__ float sdata[NWAVES];
    float acc = 0.0f;
    for (int i = threadIdx.x; i < np; i += blockDim.x) acc += partials[i];
    const float total = block_reduce(acc, sdata);
    if (threadIdx.x == 0) out[0] = total * inv_n;
}

extern "C" void kernel_launch(void* const* d_in, const int* in_sizes, int n_in,
                              void* d_out, int out_size, void* d_ws, size_t ws_size,
                              hipStream_t stream) {
    const float4* pred = (const float4*)d_in[0];
    const float4* tgt  = (const float4*)d_in[1];
    const int n = in_sizes[0] / 4;  // [N,4] flat -> N boxes

    int blocks = NBLOCKS;
    if ((size_t)blocks * sizeof(float) > ws_size)
        blocks = (int)(ws_size / sizeof(float));
    if (blocks < 1) blocks = 1;

    float* partials = (float*)d_ws;
    diou_partial_kernel<<<blocks, NTHREADS, 0, stream>>>(pred, tgt, partials, n);
    diou_final_kernel<<<1, NTHREADS, 0, stream>>>(partials, blocks,
                                                  (float*)d_out, 1.0f / (float)n);
}